// Simple_GAt_3839700762909
// MI455X (gfx1250) — compile-verified
//
#include <hip/hip_runtime.h>
#include <math.h>

#define N_NODES 10000
#define N_EDGES 320000
#define DIM     256
#define DV      128   // DIM/2

typedef __attribute__((ext_vector_type(2))) float v2f;
typedef __attribute__((ext_vector_type(8))) float v8f;

__device__ __forceinline__ float mishf(float x) {
    // x * tanh(softplus(x)), softplus = log1p(exp(x)) with overflow guard
    float sp = (x > 20.0f) ? x : log1pf(expf(x));
    return x * tanhf(sp);
}

// ---------------------------------------------------------------------------
// Kernel 1: per-node attention scores.
// s_src[n] = feat[n,:] . W_a[0:256, 0] ; s_dst[n] = feat[n,:] . W_a[256:512, 0]
// One wave (32 lanes) per node; wave32 shuffle reduction.
// ---------------------------------------------------------------------------
__global__ void __launch_bounds__(256)
node_scores_kernel(const float* __restrict__ feat,
                   const float* __restrict__ W_a,
                   float* __restrict__ s_src,
                   float* __restrict__ s_dst) {
    int node = blockIdx.x * 8 + (threadIdx.x >> 5);
    int lane = threadIdx.x & 31;
    if (node >= N_NODES) return;
    const float* f = feat + (size_t)node * DIM;
    float a0 = 0.0f, a1 = 0.0f;
#pragma unroll
    for (int i = 0; i < DIM / 32; ++i) {
        int k = lane + i * 32;
        float fv = f[k];
        a0 += fv * W_a[k];        // W_a is [512,1] row-major -> element k
        a1 += fv * W_a[DIM + k];
    }
#pragma unroll
    for (int off = 16; off > 0; off >>= 1) {
        a0 += __shfl_down(a0, off, 32);
        a1 += __shfl_down(a1, off, 32);
    }
    if (lane == 0) { s_src[node] = a0; s_dst[node] = a1; }
}

// ---------------------------------------------------------------------------
// Kernel 2: v = mish(feat @ W_v + b_v)   [N,256] x [256,128] -> [N,128]
// Block = 128 threads (4 waves). Each block owns a 16-node row tile (exact:
// 10000 = 625*16). feat tile staged in LDS; each wave computes two 16x16
// output tiles with V_WMMA_F32_16X16X4_F32 over K=256 (64 WMMAs per tile).
// No divergence around WMMA -> EXEC is all 1s as required.
// ---------------------------------------------------------------------------
__global__ void __launch_bounds__(128)
node_value_wmma_kernel(const float* __restrict__ feat,
                       const float* __restrict__ W_v,
                       const float* __restrict__ b_v,
                       float* __restrict__ vout) {
    __shared__ float lds_feat[16][DIM];   // 16 KB

    int tid   = threadIdx.x;
    int node0 = blockIdx.x * 16;

    // Stage the 16x256 feat tile: 1024 float4 / 128 threads = 8 each.
    {
        const float4* g = (const float4*)(feat + (size_t)node0 * DIM);
        float4* s = (float4*)(&lds_feat[0][0]);
#pragma unroll
        for (int i = 0; i < 8; ++i)
            s[tid + i * 128] = g[tid + i * 128];
    }
    __syncthreads();

    int wave  = tid >> 5;
    int lane  = tid & 31;
    int mrow  = lane & 15;            // A-frag row (M) / B,C,D-frag column (N)
    int khalf = (lane >> 4) * 2;      // lanes 16..31 hold K+2,K+3

#pragma unroll
    for (int t = 0; t < 2; ++t) {
        int cb  = (wave * 2 + t) * 16;   // output column base (8 tiles / block)
        int col = cb + mrow;
        v8f acc = {};
        for (int k0 = 0; k0 < DIM; k0 += 4) {
            // A 16x4 f32: VGPR0 = K {0|2}, VGPR1 = K {1|3}; rows on lanes&15
            v2f a, b;
            a.x = lds_feat[mrow][k0 + khalf];
            a.y = lds_feat[mrow][k0 + khalf + 1];
            // B 4x16 f32: row striped across lanes within a VGPR
            b.x = W_v[(size_t)(k0 + khalf) * DV + col];
            b.y = W_v[(size_t)(k0 + khalf + 1) * DV + col];
            acc = __builtin_amdgcn_wmma_f32_16x16x4_f32(
                /*neg_a=*/false, a, /*neg_b=*/false, b,
                /*c_mod=*/(short)0, acc, /*reuse_a=*/false, /*reuse_b=*/false);
        }
        // C/D layout: VGPR i -> M = i + 8*(lane>=16), N = lane&15
        int mbase = (lane >> 4) * 8;
        float bias = b_v[cb + mrow];
#pragma unroll
        for (int i = 0; i < 8; ++i) {
            float x = acc[i] + bias;
            vout[(size_t)(node0 + mbase + i) * DV + cb + mrow] = mishf(x);
        }
    }
}

// ---------------------------------------------------------------------------
// Kernel 3: edge logit + segment max.
// a_e = sigmoid(s_src[src] + s_dst[dst] + b_a); seg_max[d] = max over edges.
// sigmoid > 0, so uint bit-pattern atomicMax is order-correct (init = 0).
// ---------------------------------------------------------------------------
__global__ void __launch_bounds__(256)
edge_logit_max_kernel(const int* __restrict__ src, const int* __restrict__ dst,
                      const float* __restrict__ s_src, const float* __restrict__ s_dst,
                      const float* __restrict__ b_a,
                      float* __restrict__ ae, unsigned int* __restrict__ seg_max_u) {
    int e = blockIdx.x * blockDim.x + threadIdx.x;
    if (e >= N_EDGES) return;
    int s = src[e], d = dst[e];
    float x = s_src[s] + s_dst[d] + b_a[0];
    float a = 1.0f / (1.0f + expf(-x));
    ae[e] = a;
    atomicMax(&seg_max_u[d], __float_as_uint(a));
}

// ---------------------------------------------------------------------------
// Kernel 4: exp(a - seg_max[d]) and segment sum (f32 global atomic add).
// ---------------------------------------------------------------------------
__global__ void __launch_bounds__(256)
edge_exp_sum_kernel(const int* __restrict__ dst,
                    const float* __restrict__ seg_max,
                    float* __restrict__ ae, float* __restrict__ seg_sum) {
    int e = blockIdx.x * blockDim.x + threadIdx.x;
    if (e >= N_EDGES) return;
    int d = dst[e];
    float x = expf(ae[e] - seg_max[d]);
    ae[e] = x;
    atomicAdd(&seg_sum[d], x);
}

// ---------------------------------------------------------------------------
// Kernel 5: gather-multiply-scatter. One wave per edge; 128 f32 per edge,
// 4 per lane. v (5 MB) and out (5 MB) are L2-resident on MI455X (192 MB L2).
// ---------------------------------------------------------------------------
__global__ void __launch_bounds__(256)
edge_scatter_kernel(const int* __restrict__ src, const int* __restrict__ dst,
                    const float* __restrict__ ae, const float* __restrict__ seg_sum,
                    const float* __restrict__ v, float* __restrict__ out) {
    int e    = blockIdx.x * 8 + (threadIdx.x >> 5);
    int lane = threadIdx.x & 31;
    if (e >= N_EDGES) return;
    int s = src[e], d = dst[e];
    float c = ae[e] / seg_sum[d];
    const float* vs = v + (size_t)s * DV;
    float* od = out + (size_t)d * DV;
#pragma unroll
    for (int i = 0; i < 4; ++i) {
        int j = lane + i * 32;
        atomicAdd(&od[j], vs[j] * c);
    }
}

// ---------------------------------------------------------------------------
extern "C" void kernel_launch(void* const* d_in, const int* in_sizes, int n_in,
                              void* d_out, int out_size, void* d_ws, size_t ws_size,
                              hipStream_t stream) {
    const float* feat = (const float*)d_in[0];
    const float* W_a  = (const float*)d_in[1];
    const float* b_a  = (const float*)d_in[2];
    const float* W_v  = (const float*)d_in[3];
    const float* b_v  = (const float*)d_in[4];
    const int*   src  = (const int*)d_in[5];
    const int*   dst  = (const int*)d_in[6];
    float* out = (float*)d_out;

    // Workspace carve-up (all f32): 4*N + E + N*128 = 6.56 MB
    float* s_src   = (float*)d_ws;
    float* s_dst   = s_src + N_NODES;
    float* seg_max = s_dst + N_NODES;
    float* seg_sum = seg_max + N_NODES;
    float* ae      = seg_sum + N_NODES;
    float* v       = ae + N_EDGES;

    // seg_max (as +0.0f) and seg_sum to zero; output accumulator to zero.
    hipMemsetAsync(seg_max, 0, 2 * (size_t)N_NODES * sizeof(float), stream);
    hipMemsetAsync(d_out, 0, (size_t)out_size * sizeof(float), stream);

    node_scores_kernel<<<N_NODES / 8, 256, 0, stream>>>(feat, W_a, s_src, s_dst);
    node_value_wmma_kernel<<<N_NODES / 16, 128, 0, stream>>>(feat, W_v, b_v, v);
    edge_logit_max_kernel<<<(N_EDGES + 255) / 256, 256, 0, stream>>>(
        src, dst, s_src, s_dst, b_a, ae, (unsigned int*)seg_max);
    edge_exp_sum_kernel<<<(N_EDGES + 255) / 256, 256, 0, stream>>>(
        dst, seg_max, ae, seg_sum);
    edge_scatter_kernel<<<(N_EDGES + 7) / 8, 256, 0, stream>>>(
        src, dst, ae, seg_sum, v, out);
}